// VariationalLSTM_48576080118500
// MI455X (gfx1250) — compile-verified
//
#include <hip/hip_runtime.h>
#include <hip/hip_bf16.h>

typedef float v2f __attribute__((ext_vector_type(2)));
typedef float v8f __attribute__((ext_vector_type(8)));

#define T_STEPS 512
#define BATCH   64
#define HID     512
#define LDS_STRIDE 516  // 512 + 4 pad -> conflict-free ds_load_b64 across 32 lanes

__device__ __forceinline__ float sigf(float x) { return 1.0f / (1.0f + __expf(-x)); }

// device-scope (L2-coherent) 8-byte load of two consecutive floats
__device__ __forceinline__ v2f ld_dev_f2(const float* p) {
  unsigned long long u =
      __hip_atomic_load((unsigned long long*)p, __ATOMIC_RELAXED, __HIP_MEMORY_SCOPE_AGENT);
  union { unsigned long long u; v2f f; } cv;
  cv.u = u;
  return cv.f;
}
__device__ __forceinline__ void st_dev_f(float* p, float v) {
  __hip_atomic_store(p, v, __ATOMIC_RELAXED, __HIP_MEMORY_SCOPE_AGENT);
}

// monotonic-counter grid-wide barrier (all WGs co-resident: grid = 64 WGs)
__device__ __forceinline__ void grid_sync(unsigned* cnt, unsigned target) {
  __threadfence();
  __syncthreads();
  if (threadIdx.x == 0) {
    __hip_atomic_fetch_add(cnt, 1u, __ATOMIC_ACQ_REL, __HIP_MEMORY_SCOPE_AGENT);
    while (__hip_atomic_load(cnt, __ATOMIC_ACQUIRE, __HIP_MEMORY_SCOPE_AGENT) < target) {
      __builtin_amdgcn_s_sleep(1);
    }
  }
  __syncthreads();
}

// One bidirectional LSTM layer, persistent over all T timesteps.
// grid = (32 column-WGs, 2 directions), block = 128 (4 waves of 32).
// Each WG owns h columns [16*bx, 16*bx+16); per step it computes the 4 gate
// tiles for those columns (fused x@Wih^T + h@Whh^T via V_WMMA_F32_16X16X4_F32),
// applies the cell nonlinearity in registers, keeps c/h carry in registers.
// The WG's Whh slab (64 gate-rows x 512) is staged in LDS once and reused
// for all 512 timesteps by all 4 waves.
__global__ __launch_bounds__(128, 1)
void lstm_layer_kernel(const float* __restrict__ xin, int in_dim,
                       const int* __restrict__ lengths,
                       const float* __restrict__ Wih_f, const float* __restrict__ Whh_f,
                       const float* __restrict__ b0_f,  const float* __restrict__ b1_f,
                       const float* __restrict__ Wih_b, const float* __restrict__ Whh_b,
                       const float* __restrict__ b0_b,  const float* __restrict__ b1_b,
                       float* __restrict__ outs,  // this layer's out section [T][B][2H]
                       float* __restrict__ hns, float* __restrict__ cns,
                       float* __restrict__ hbuf,  // [2 dir][2 pingpong][B][H]
                       unsigned* __restrict__ counter, int layer) {
  __shared__ float sWhh[64 * LDS_STRIDE];  // ~129 KB (CDNA5: 320 KB LDS per WGP)

  const int dir = blockIdx.y;  // 0 = forward, 1 = backward
  const float* Wih = dir ? Wih_b : Wih_f;
  const float* Whh = dir ? Whh_b : Whh_f;
  const float* b0  = dir ? b0_b  : b0_f;
  const float* b1  = dir ? b1_b  : b1_f;

  const int wave  = threadIdx.x >> 5;  // M-tile (batch rows 16*wave..)
  const int lane  = threadIdx.x & 31;
  const int lhalf = lane >> 4;
  const int lr    = lane & 15;
  const int colbase = blockIdx.x * 16;       // h-column base within H
  const int am    = wave * 16 + lr;          // A-fragment batch row
  const int ncol  = colbase + lr;            // acc/B column within H
  const int kboff = 2 * lhalf;               // k sub-offset within 4-slice
  const int r0    = wave * 16 + lhalf * 8;   // first acc row owned by this lane

  // ---- stage this WG's Whh slab (64 gate-rows x 512) into LDS, once ----
  for (int idx = threadIdx.x; idx < 64 * HID; idx += 128) {
    int rr = idx >> 9;        // slab row 0..63  (gate = rr/16, col = rr%16)
    int cc = idx & (HID - 1); // k column
    int ng = (rr >> 4) * HID + colbase + (rr & 15);
    sWhh[rr * LDS_STRIDE + cc] = Whh[(size_t)ng * HID + cc];
  }
  __syncthreads();

  // Per-gate B pointers for the x-part (constant over time) and fused bias
  const float* bXp0; const float* bXp1; const float* bXp2; const float* bXp3;
  float bias[4];
  {
    int n0 = 0 * HID + ncol, n1 = 1 * HID + ncol, n2 = 2 * HID + ncol, n3 = 3 * HID + ncol;
    bXp0 = Wih + (size_t)n0 * in_dim + kboff;
    bXp1 = Wih + (size_t)n1 * in_dim + kboff;
    bXp2 = Wih + (size_t)n2 * in_dim + kboff;
    bXp3 = Wih + (size_t)n3 * in_dim + kboff;
    bias[0] = b0[n0] + b1[n0]; bias[1] = b0[n1] + b1[n1];
    bias[2] = b0[n2] + b1[n2]; bias[3] = b0[n3] + b1[n3];
  }
  // LDS offsets of this lane's Whh B-fragment rows, per gate
  const int bo0 = (0 * 16 + lr) * LDS_STRIDE + kboff;
  const int bo1 = (1 * 16 + lr) * LDS_STRIDE + kboff;
  const int bo2 = (2 * 16 + lr) * LDS_STRIDE + kboff;
  const int bo3 = (3 * 16 + lr) * LDS_STRIDE + kboff;

  int   lenv[8];
  float creg[8], hreg[8];
#pragma unroll
  for (int v = 0; v < 8; ++v) { lenv[v] = lengths[r0 + v]; creg[v] = 0.f; hreg[v] = 0.f; }

  float* hd = hbuf + (size_t)dir * (2 * BATCH * HID);
  const unsigned nwg = gridDim.x * gridDim.y;

  for (int s = 0; s < T_STEPS; ++s) {
    const int t = dir ? (T_STEPS - 1 - s) : s;
    const float* hcur = hd + (s & 1) * (BATCH * HID);
    float*       hnxt = hd + ((s + 1) & 1) * (BATCH * HID);

    v8f acc0 = {}, acc1 = {}, acc2 = {}, acc3 = {};

    // prefetch next timestep's x row into caches
    {
      int tn = dir ? (t - 1) : (t + 1);
      if (tn >= 0 && tn < T_STEPS)
        __builtin_prefetch(xin + ((size_t)tn * BATCH + am) * in_dim, 0, 1);
    }

    // ---- input contribution: x_t @ Wih^T (global, streams through L2) ----
    const float* aX = xin + ((size_t)t * BATCH + am) * in_dim + kboff;
#pragma unroll 8
    for (int k = 0; k < in_dim; k += 4) {
      v2f a  = *(const v2f*)(aX + k);
      v2f q0 = *(const v2f*)(bXp0 + k);
      v2f q1 = *(const v2f*)(bXp1 + k);
      v2f q2 = *(const v2f*)(bXp2 + k);
      v2f q3 = *(const v2f*)(bXp3 + k);
      acc0 = __builtin_amdgcn_wmma_f32_16x16x4_f32(false, a, false, q0, (short)0, acc0, false, false);
      acc1 = __builtin_amdgcn_wmma_f32_16x16x4_f32(false, a, false, q1, (short)0, acc1, false, false);
      acc2 = __builtin_amdgcn_wmma_f32_16x16x4_f32(false, a, false, q2, (short)0, acc2, false, false);
      acc3 = __builtin_amdgcn_wmma_f32_16x16x4_f32(false, a, false, q3, (short)0, acc3, false, false);
    }

    // ---- recurrent contribution: h_{t-1} @ Whh^T ----
    // A: device-scope h reads (cross-WGP coherent); B: Whh from LDS.
    const float* aH = hcur + (size_t)am * HID + kboff;
#pragma unroll 8
    for (int k = 0; k < HID; k += 4) {
      v2f a  = ld_dev_f2(aH + k);
      v2f q0 = *(const v2f*)(sWhh + bo0 + k);
      v2f q1 = *(const v2f*)(sWhh + bo1 + k);
      v2f q2 = *(const v2f*)(sWhh + bo2 + k);
      v2f q3 = *(const v2f*)(sWhh + bo3 + k);
      acc0 = __builtin_amdgcn_wmma_f32_16x16x4_f32(false, a, false, q0, (short)0, acc0, false, false);
      acc1 = __builtin_amdgcn_wmma_f32_16x16x4_f32(false, a, false, q1, (short)0, acc1, false, false);
      acc2 = __builtin_amdgcn_wmma_f32_16x16x4_f32(false, a, false, q2, (short)0, acc2, false, false);
      acc3 = __builtin_amdgcn_wmma_f32_16x16x4_f32(false, a, false, q3, (short)0, acc3, false, false);
    }

    // ---- LSTM cell, fully in registers (all 4 accs share the same layout) ----
    float* ocol = outs + (size_t)t * BATCH * (2 * HID) + (size_t)dir * HID + ncol;
#pragma unroll
    for (int v = 0; v < 8; ++v) {
      const int row = r0 + v;
      float gi = acc0[v] + bias[0];
      float gf = acc1[v] + bias[1];
      float gg = acc2[v] + bias[2];
      float go = acc3[v] + bias[3];
      float cn = sigf(gf) * creg[v] + sigf(gi) * tanhf(gg);
      float hn = sigf(go) * tanhf(cn);
      bool  m  = t < lenv[v];
      float hk = m ? hn : hreg[v];
      float ck = m ? cn : creg[v];
      hreg[v] = hk;
      creg[v] = ck;
      st_dev_f(hnxt + (size_t)row * HID + ncol, hk);
      ocol[(size_t)row * (2 * HID)] = m ? hn : 0.0f;
    }

    grid_sync(counter, nwg * (unsigned)(s + 1));
  }

  // final carried states -> hns / cns  (order: l0f, l0b, l1f, l1b)
  size_t fb = ((size_t)(layer * 2 + dir) * BATCH + r0) * HID + ncol;
#pragma unroll
  for (int v = 0; v < 8; ++v) {
    hns[fb + (size_t)v * HID] = hreg[v];
    cns[fb + (size_t)v * HID] = creg[v];
  }
}

extern "C" void kernel_launch(void* const* d_in, const int* in_sizes, int n_in,
                              void* d_out, int out_size, void* d_ws, size_t ws_size,
                              hipStream_t stream) {
  const float* x       = (const float*)d_in[0];
  const int*   lengths = (const int*)d_in[1];

  const float* P[16];
  for (int i = 0; i < 16 && (2 + i) < n_in; ++i) P[i] = (const float*)d_in[2 + i];

  // Disambiguate pytree flatten order using layer-1 sizes:
  // Wih(l1) = 2048*1024, Whh = 2048*512. Sorted-dict order => Whh first, 'b' dir first.
  bool insertion = (n_in > 10) && (in_sizes[10] == 2048 * 1024);

  const float* Wih[2][2]; const float* Whh[2][2];  // [layer][dir: 0=f,1=b]
  const float* Ba[2][2];  const float* Bb[2][2];
  for (int l = 0; l < 2; ++l) {
    const float* const* Bk = P + 8 * l;
    if (insertion) {  // [f.Wih, f.Whh, f.bih, f.bhh, b.Wih, b.Whh, b.bih, b.bhh]
      Wih[l][0] = Bk[0]; Whh[l][0] = Bk[1]; Ba[l][0] = Bk[2]; Bb[l][0] = Bk[3];
      Wih[l][1] = Bk[4]; Whh[l][1] = Bk[5]; Ba[l][1] = Bk[6]; Bb[l][1] = Bk[7];
    } else {          // sorted: [b.Whh, b.Wih, b.bhh, b.bih, f.Whh, f.Wih, f.bhh, f.bih]
      Whh[l][1] = Bk[0]; Wih[l][1] = Bk[1]; Ba[l][1] = Bk[2]; Bb[l][1] = Bk[3];
      Whh[l][0] = Bk[4]; Wih[l][0] = Bk[5]; Ba[l][0] = Bk[6]; Bb[l][0] = Bk[7];
    }
  }

  float* out = (float*)d_out;
  const size_t OUTS_PER_LAYER = (size_t)T_STEPS * BATCH * 2 * HID;  // 33,554,432
  float* hns = out + 2 * OUTS_PER_LAYER;
  float* cns = hns + (size_t)4 * BATCH * HID;

  unsigned* counter = (unsigned*)d_ws;
  float* hbuf = (float*)((char*)d_ws + 1024);
  size_t wsneed = 1024 + (size_t)2 * 2 * BATCH * HID * sizeof(float);  // ~525 KB

  dim3 grid(32, 2), block(128);

  // layer 0: input = x (in_dim = 512), outs -> section 0 of d_out
  hipMemsetAsync(d_ws, 0, wsneed, stream);
  lstm_layer_kernel<<<grid, block, 0, stream>>>(
      x, 512, lengths,
      Wih[0][0], Whh[0][0], Ba[0][0], Bb[0][0],
      Wih[0][1], Whh[0][1], Ba[0][1], Bb[0][1],
      out, hns, cns, hbuf, counter, 0);

  // layer 1: input = layer-0 output section (in_dim = 1024), outs -> section 1
  hipMemsetAsync(d_ws, 0, wsneed, stream);
  lstm_layer_kernel<<<grid, block, 0, stream>>>(
      out, 1024, lengths,
      Wih[1][0], Whh[1][0], Ba[1][0], Bb[1][0],
      Wih[1][1], Whh[1][1], Ba[1][1], Bb[1][1],
      out + OUTS_PER_LAYER, hns, cns, hbuf, counter, 1);
}